// GraphConv4D_20289425506359
// MI455X (gfx1250) — compile-verified
//
#include <hip/hip_runtime.h>

// EdgeConv (DGCNN GraphConv4D) for MI455X / gfx1250, wave32, WMMA f32 16x16x4.
// B=8, C=64, N=4096, K=16, OUT=64.

typedef float v8f __attribute__((ext_vector_type(8)));
typedef float v2f __attribute__((ext_vector_type(2)));

#define Bn 8
#define Cn 64
#define Nn 4096
#define Kn 16
#define On 64

__global__ __launch_bounds__(256) void edgeconv_wmma_kernel(
    const float* __restrict__ x,     // (B, C, N)
    const int*   __restrict__ edge,  // (2, B, N, K) int32
    const float* __restrict__ W,     // (OUT, 2C)
    const float* __restrict__ bias,  // (OUT,)
    float*       __restrict__ out)   // (B, OUT, N)
{
    // Reparameterized weights Wp[o][c]: c<64 -> W1-W2, c>=64 -> W2
    __shared__ float ldsW[On * 2 * Cn];  // 8192 floats = 32 KB
    __shared__ float ldsB[On];

    const int tid = threadIdx.x;

    // Cooperative, branch-free load of Wp into LDS (32 elements/thread).
    // W[o*128 + (c|64)] aliases W[o*128+c] when c>=64, so the subtrahend is a
    // plain select (v_cndmask), no divergent branch -> loop unrolls cleanly.
#pragma unroll
    for (int i = tid; i < On * 2 * Cn; i += 256) {
        const int o = i >> 7;        // /128
        const int c = i & 127;
        const float w  = W[o * 128 + c];
        const float w2 = W[o * 128 + (c | Cn)];
        ldsW[o * 128 + c] = w - ((c < Cn) ? w2 : 0.0f);
    }
    if (tid < On) ldsB[tid] = bias[tid];
    __syncthreads();

    const int lane = tid & 31;
    const int wave = tid >> 5;
    const int p  = blockIdx.x * 8 + wave;   // point id, 0..B*N-1
    const int bb = p >> 12;                 // /N
    const int n  = p & (Nn - 1);

    const int col  = lane & 15;             // which of the K=16 neighbors (B/N column)
    const int half = lane >> 4;             // lane half selects K-substep (+2) / M (+8)

    // Neighbor indices for this lane's column.
    const int* eb = edge + (size_t)bb * (Nn * Kn) + (size_t)n * Kn + col;
    const int mj = eb[0];                           // edge_index[0] : x_j gather
    const int mi = eb[(size_t)Bn * Nn * Kn];        // edge_index[1] : x_i gather

    // Per-lane gather bases: channel offset starts at 2*half (B-fragment layout).
    const float* xb = x + (size_t)bb * Cn * Nn + (size_t)(2 * half) * Nn;
    const float* pi = xb + mi;
    const float* pj = xb + mj;

    // Gather all 32 B-fragments once (64 VGPRs), reused by all 4 A-tiles.
    // B frag s, component v, lane l  ==  h[c = 4s + v + 2*half][col]
    v2f bf[32];
#pragma unroll
    for (int s = 0; s < 16; ++s) {
        bf[s][0]      = pi[(4 * s)     * Nn];   // x_i channels 0..63
        bf[s][1]      = pi[(4 * s + 1) * Nn];
        bf[16 + s][0] = pj[(4 * s)     * Nn];   // x_j channels 0..63 (Wp cols 64..127)
        bf[16 + s][1] = pj[(4 * s + 1) * Nn];
    }

    float* outp = out + (size_t)bb * On * Nn + n;

#pragma unroll
    for (int m = 0; m < 4; ++m) {   // 4 tiles of 16 output channels
        // Two independent accumulation chains per tile (even/odd K-steps) for
        // back-to-back WMMA issue; chain0 seeded with bias, chain1 with zero.
        // C/D layout: VGPR v, lane-half -> M = 16m + v + 8*half.
        v8f acc0, acc1;
#pragma unroll
        for (int v = 0; v < 8; ++v) { acc0[v] = ldsB[16 * m + 8 * half + v]; acc1[v] = 0.0f; }

        // A frag (m, s), component v, lane l == Wp[16m + col][4s + v + 2*half]
        const float* wrow = &ldsW[(16 * m + col) * 128 + 2 * half];
#pragma unroll
        for (int s = 0; s < 32; s += 2) {
            v2f a0 = *(const v2f*)(wrow + 4 * s);        // 8B-aligned LDS loads
            v2f a1 = *(const v2f*)(wrow + 4 * (s + 1));
            acc0 = __builtin_amdgcn_wmma_f32_16x16x4_f32(
                false, a0, false, bf[s],     (short)0, acc0, false, false);
            acc1 = __builtin_amdgcn_wmma_f32_16x16x4_f32(
                false, a1, false, bf[s + 1], (short)0, acc1, false, false);
        }

        // max over K=16 neighbors = max over the 16 columns (lanes within a half),
        // then ReLU (commutes with max), then scatter 16 channel results.
#pragma unroll
        for (int v = 0; v < 8; ++v) {
            float mv = acc0[v] + acc1[v];
            mv = fmaxf(mv, __shfl_xor(mv, 1, 32));
            mv = fmaxf(mv, __shfl_xor(mv, 2, 32));
            mv = fmaxf(mv, __shfl_xor(mv, 4, 32));
            mv = fmaxf(mv, __shfl_xor(mv, 8, 32));
            mv = fmaxf(mv, 0.0f);
            if (col == v) outp[(size_t)(16 * m + 8 * half + v) * Nn] = mv;
        }
    }
}

extern "C" void kernel_launch(void* const* d_in, const int* in_sizes, int n_in,
                              void* d_out, int out_size, void* d_ws, size_t ws_size,
                              hipStream_t stream) {
    const float* x    = (const float*)d_in[0];  // (B,C,N,1)
    const int*   edge = (const int*)d_in[1];    // (2,B,N,K) int32
    const float* W    = (const float*)d_in[2];  // (OUT,2C)
    const float* b    = (const float*)d_in[3];  // (OUT,)
    float*       out  = (float*)d_out;          // (B,OUT,N,1)

    dim3 block(256);
    dim3 grid((Bn * Nn) / 8);  // 8 waves per block, 1 point per wave
    edgeconv_wmma_kernel<<<grid, block, 0, stream>>>(x, edge, W, b, out);
}